// AlphaKnotLoss_6141803233392
// MI455X (gfx1250) — compile-verified
//
#include <hip/hip_runtime.h>
#include <hip/hip_bf16.h>

// ---------------------------------------------------------------------------
// AlphaKnot loss: memory-bound streaming reduction.
//   per-graph: S1 = sum(target_probs), S2 = sum(target_probs*logits),
//              S3 = sum(exp(logits)) over the flat segment (node dim folds in).
//   loss = mean((values-target_vals)^2) + mean((log(S3+eps)-S2)/(S1+eps))
// Roofline: 168 MB @ 23.3 TB/s ~= 7.2 us; compute negligible -> stream once
// with b128 loads, wave32 WMMA-based lane reduction (CDNA5 matrix path).
// ---------------------------------------------------------------------------

#define EPSF 1e-9f

typedef float v2f __attribute__((ext_vector_type(2)));
typedef float v8f __attribute__((ext_vector_type(8)));

// Half-wave partial sum with the contract:
//   result(lane 0) + result(lane 16) == sum of v over all 32 lanes.
// WMMA path: V_WMMA_F32_16X16X4_F32, A = lane values in K-slot 0 (a.y = 0
// zeroes K=1/3), B = ones  =>  D[m][n] = v[m] + v[16+m].  Summing the 8 D
// VGPRs gives lanes 0-15: sum(v[0..7],v[16..23]); lanes 16-31: the rest.
__device__ __forceinline__ float wave_sum_halves(float v) {
#if __has_builtin(__builtin_amdgcn_wmma_f32_16x16x4_f32)
    v2f a; a.x = v;    a.y = 0.0f;
    v2f b; b.x = 1.0f; b.y = 1.0f;
    v8f c = {};
    c = __builtin_amdgcn_wmma_f32_16x16x4_f32(
        /*neg_a=*/false, a, /*neg_b=*/false, b,
        /*c_mod=*/(short)0, c, /*reuse_a=*/false, /*reuse_b=*/false);
    return ((c[0] + c[1]) + (c[2] + c[3])) + ((c[4] + c[5]) + (c[6] + c[7]));
#else
    // Reduce within each 16-lane half: lane0 = sum(0..15), lane16 = sum(16..31).
    #pragma unroll
    for (int off = 8; off > 0; off >>= 1) v += __shfl_xor(v, off, 32);
    return v;
#endif
}

// ---------------------------------------------------------------------------
// Kernel 1: exclusive prefix sum of batch_counts -> offs[0..B].
// Wave-level shfl_up scans (no intra-wave barriers) + one wave-0 scan of the
// 32 wave totals: 4 block barriers per 4096-element chunk.
// ---------------------------------------------------------------------------
__global__ __launch_bounds__(1024)
void ak_scan_offsets(const int* __restrict__ counts, int* __restrict__ offs, int B) {
    __shared__ int wtot[32];   // inclusive per-wave totals
    __shared__ int woff[32];   // exclusive per-wave offsets within chunk
    __shared__ int carry_sh;
    const int tid  = threadIdx.x;
    const int lane = tid & 31;
    const int wid  = tid >> 5;
    if (tid == 0) carry_sh = 0;
    __syncthreads();

    for (int base = 0; base < B; base += 4 * 1024) {
        const int idx = base + tid * 4;
        const int c0 = (idx + 0 < B) ? counts[idx + 0] : 0;
        const int c1 = (idx + 1 < B) ? counts[idx + 1] : 0;
        const int c2 = (idx + 2 < B) ? counts[idx + 2] : 0;
        const int c3 = (idx + 3 < B) ? counts[idx + 3] : 0;
        const int s  = c0 + c1 + c2 + c3;

        // Intra-wave inclusive scan of per-thread sums.
        int incl = s;
        #pragma unroll
        for (int off = 1; off < 32; off <<= 1) {
            const int t = __shfl_up(incl, off, 32);
            if (lane >= off) incl += t;
        }
        if (lane == 31) wtot[wid] = incl;
        __syncthreads();

        // Wave 0 scans the 32 wave totals.
        if (wid == 0) {
            const int w = wtot[lane];
            int wi = w;
            #pragma unroll
            for (int off = 1; off < 32; off <<= 1) {
                const int t = __shfl_up(wi, off, 32);
                if (lane >= off) wi += t;
            }
            woff[lane] = wi - w;   // exclusive offset of wave `lane`
        }
        __syncthreads();

        const int carry = carry_sh;
        const int excl  = carry + woff[wid] + (incl - s);
        if (idx + 0 < B) offs[idx + 0] = excl;
        if (idx + 1 < B) offs[idx + 1] = excl + c0;
        if (idx + 2 < B) offs[idx + 2] = excl + c0 + c1;
        if (idx + 3 < B) offs[idx + 3] = excl + c0 + c1 + c2;
        __syncthreads();
        if (tid == 0) carry_sh = carry + woff[31] + wtot[31];
        __syncthreads();
    }
    if (tid == 0) offs[B] = carry_sh;
}

// ---------------------------------------------------------------------------
// Kernel 2: one block per graph; stream segment with b128 loads.
// Segment starts at node*10 floats = 40B boundary: 16B-aligned when the start
// node is even, else 8B-aligned -> peel a 2-float head, b128 body, 2-float
// tail. (Common case count=512: head = tail = 0.)
// ---------------------------------------------------------------------------
__global__ __launch_bounds__(256)
void ak_graph_loss(const float* __restrict__ logits,
                   const float* __restrict__ tprobs,
                   const int*   __restrict__ offs,
                   float*       __restrict__ lp) {
    const int  g     = blockIdx.x;
    const int  start = offs[g];
    const int  end   = offs[g + 1];
    const long base  = (long)start * 10;           // float index (always even)
    const long n     = (long)(end - start) * 10;   // floats in segment (even)

    long head = (base & 3) ? 2 : 0;                // floats to 16B alignment
    if (head > n) head = n;
    const long rem  = n - head;
    const long n4   = rem >> 2;                    // b128 body
    const int  tail = (int)(rem & 3);              // 0 or 2 floats

    const float4* __restrict__ l4 = (const float4*)(logits + base + head);
    const float4* __restrict__ p4 = (const float4*)(tprobs + base + head);

    float s1 = 0.0f, s2 = 0.0f, s3 = 0.0f;
    #pragma unroll 2
    for (long i = threadIdx.x; i < n4; i += 256) {
        const float4 p = p4[i];
        const float4 l = l4[i];
        s1 += (p.x + p.y) + (p.z + p.w);
        s2 = fmaf(p.x, l.x, fmaf(p.y, l.y, fmaf(p.z, l.z, fmaf(p.w, l.w, s2))));
        s3 += (__expf(l.x) + __expf(l.y)) + (__expf(l.z) + __expf(l.w));
    }
    if (threadIdx.x == 0) {
        if (head) {
            const float2 p = *(const float2*)(tprobs + base);
            const float2 l = *(const float2*)(logits + base);
            s1 += p.x + p.y;
            s2 = fmaf(p.x, l.x, fmaf(p.y, l.y, s2));
            s3 += __expf(l.x) + __expf(l.y);
        }
        if (tail) {
            const long t = base + head + 4 * n4;
            const float2 p = *(const float2*)(tprobs + t);
            const float2 l = *(const float2*)(logits + t);
            s1 += p.x + p.y;
            s2 = fmaf(p.x, l.x, fmaf(p.y, l.y, s2));
            s3 += __expf(l.x) + __expf(l.y);
        }
    }

    // Wave-level reduction (reconverged: EXEC all ones for WMMA).
    const float t1 = wave_sum_halves(s1);
    const float t2 = wave_sum_halves(s2);
    const float t3 = wave_sum_halves(s3);

    __shared__ float r1[16], r2[16], r3[16];
    const int lane = threadIdx.x & 31;
    const int wid  = threadIdx.x >> 5;
    if (lane == 0 || lane == 16) {
        const int slot = (wid << 1) | (lane >> 4);
        r1[slot] = t1; r2[slot] = t2; r3[slot] = t3;
    }
    __syncthreads();
    if (threadIdx.x == 0) {
        float S1 = 0.0f, S2 = 0.0f, S3 = 0.0f;
        #pragma unroll
        for (int w = 0; w < 16; ++w) { S1 += r1[w]; S2 += r2[w]; S3 += r3[w]; }
        lp[g] = (__logf(S3 + EPSF) - S2) / (S1 + EPSF);
    }
}

// ---------------------------------------------------------------------------
// Kernel 3: mean of per-graph losses + value MSE -> out[0]  (1 block / 256)
// ---------------------------------------------------------------------------
__global__ __launch_bounds__(256)
void ak_finalize(const float* __restrict__ values,
                 const float* __restrict__ tvals,
                 const float* __restrict__ lp,
                 float*       __restrict__ out,
                 int B) {
    float accL = 0.0f, accM = 0.0f;
    for (int i = threadIdx.x; i < B; i += blockDim.x) {
        accL += lp[i];
        const float d = values[i] - tvals[i];
        accM = fmaf(d, d, accM);
    }
    const float tL = wave_sum_halves(accL);
    const float tM = wave_sum_halves(accM);

    __shared__ float rL[16], rM[16];
    const int lane = threadIdx.x & 31;
    const int wid  = threadIdx.x >> 5;
    if (lane == 0 || lane == 16) {
        const int slot = (wid << 1) | (lane >> 4);
        rL[slot] = tL; rM[slot] = tM;
    }
    __syncthreads();
    if (threadIdx.x == 0) {
        float L = 0.0f, M = 0.0f;
        #pragma unroll
        for (int w = 0; w < 16; ++w) { L += rL[w]; M += rM[w]; }
        const float invB = 1.0f / (float)B;
        out[0] = M * invB + L * invB;
    }
}

// ---------------------------------------------------------------------------
extern "C" void kernel_launch(void* const* d_in, const int* in_sizes, int n_in,
                              void* d_out, int out_size, void* d_ws, size_t ws_size,
                              hipStream_t stream) {
    const float* logits  = (const float*)d_in[0];
    const float* values  = (const float*)d_in[1];
    const float* tprobs  = (const float*)d_in[2];
    const float* tvals   = (const float*)d_in[3];
    const int*   counts  = (const int*)d_in[4];
    float*       out     = (float*)d_out;

    const int B = in_sizes[1];   // number of graphs

    // Workspace layout: [ int offs[B+1] | pad to 256B | float lp[B] ]
    int*   offs = (int*)d_ws;
    size_t lp_off = (((size_t)(B + 1) * sizeof(int)) + 255) & ~(size_t)255;
    float* lp   = (float*)((char*)d_ws + lp_off);

    ak_scan_offsets<<<1, 1024, 0, stream>>>(counts, offs, B);
    ak_graph_loss  <<<B, 256, 0, stream>>>(logits, tprobs, offs, lp);
    ak_finalize    <<<1, 256, 0, stream>>>(values, tvals, lp, out, B);
}